// GuidedMatching_78082505441393
// MI455X (gfx1250) — compile-verified
//
#include <hip/hip_runtime.h>

// ---------------------------------------------------------------------------
// Guided matching (LoFTR-style). Memory/atomic-bound; no matmul -> no WMMA
// (by design: zero MAC structure in this op).
// Phase A: zero per-cell winner tables (u64 packed (scoreKey<<32)|~idx), B128.
// Phase B: segmented top-1 per cell via GLOBAL_ATOMIC_MAX_U64 (no-return),
//          streaming inputs loaded with TH=NT so L2 keeps the gather tables.
// Phase C: per-match lookup + gather + scale + mask; NT stores for outputs.
// ---------------------------------------------------------------------------

typedef float v2f __attribute__((ext_vector_type(2)));

static __device__ __forceinline__ v2f nt_load_v2f(const float* p, long long i) {
    return __builtin_nontemporal_load(((const v2f*)p) + i);   // global_load_b64 TH_NT
}
static __device__ __forceinline__ float nt_load_f(const float* p, long long i) {
    return __builtin_nontemporal_load(p + i);
}
static __device__ __forceinline__ int nt_load_i(const int* p, long long i) {
    return __builtin_nontemporal_load(p + i);
}

static __device__ __forceinline__ unsigned score_key(float s) {
    // Monotonic (totally ordered) mapping of float bits to unsigned.
    unsigned u = __float_as_uint(s);
    return (u & 0x80000000u) ? ~u : (u | 0x80000000u);
}

// Phase A: zero both winner tables; num_cells = B*hc*wc read on device.
__global__ void gm_init_kernel(unsigned long long* __restrict__ win,
                               const int* __restrict__ pB,
                               const int* __restrict__ phc,
                               const int* __restrict__ pwc) {
    const long long total = 2LL * (long long)pB[0] * (long long)phc[0] * (long long)pwc[0];
    const long long pairs = total >> 1;
    const long long stride = (long long)gridDim.x * (long long)blockDim.x;
    ulonglong2* __restrict__ win2 = (ulonglong2*)win;
    const ulonglong2 z2 = {0ULL, 0ULL};
    for (long long i = (long long)blockIdx.x * blockDim.x + threadIdx.x; i < pairs; i += stride)
        win2[i] = z2;                                          // global_store_b128
    // tail (total odd) — cannot happen for real shapes, kept for generality
    if ((total & 1LL) && blockIdx.x == 0 && threadIdx.x == 0)
        win[total - 1] = 0ULL;
}

// Phase B: top-1 (max score, min index) per cell for one detector set.
// win table for this set starts at winBase + part * (B*hc*wc).
__global__ void gm_top1_kernel(const float* __restrict__ kpts,    // n x 2
                               const float* __restrict__ scores,  // n
                               const int*   __restrict__ bids,    // n
                               int n, int part,
                               unsigned long long* __restrict__ winBase,
                               const int* __restrict__ pB,
                               const int* __restrict__ phc,
                               const int* __restrict__ pwc,
                               const int* __restrict__ phi) {
    const int hc = phc[0], wc = pwc[0];
    const float s = (float)((double)hc / (double)phi[0]);   // s_i2c = hc/hi
    const long long nc = (long long)pB[0] * hc * wc;
    unsigned long long* __restrict__ win = winBase + (long long)part * nc;

    int i = blockIdx.x * blockDim.x + threadIdx.x;
    if (i >= n) return;

    v2f kp = nt_load_v2f(kpts, i);                         // single-use stream: TH_NT
    int cx = (int)rintf(kp.x * s);                         // round-half-even
    int cy = (int)rintf(kp.y * s);
    cx = min(max(cx, 0), wc - 1);
    cy = min(max(cy, 0), hc - 1);
    int cell = nt_load_i(bids, i) * (hc * wc) + cy * wc + cx;

    unsigned long long packed =
        ((unsigned long long)score_key(nt_load_f(scores, i)) << 32) |
        (unsigned long long)(~(unsigned)i);                // ~idx: ties -> min idx
    atomicMax(&win[cell], packed);                         // GLOBAL_ATOMIC_MAX_U64, no return
}

// Phase C: per-match winner lookup + gather + scale + validity mask.
__global__ void gm_match_kernel(const float* __restrict__ mk0,    // M x 2
                                const float* __restrict__ mk1,    // M x 2
                                const float* __restrict__ mconf,  // M
                                const float* __restrict__ dk0,    // M0 x 2
                                const float* __restrict__ dk1,    // M1 x 2
                                const float* __restrict__ scale0, // B x 2
                                const float* __restrict__ scale1, // B x 2
                                const int*   __restrict__ mbids,  // M
                                const unsigned long long* __restrict__ winBase,
                                int M, unsigned M0, unsigned M1,
                                const int* __restrict__ pB,
                                const int* __restrict__ phc,
                                const int* __restrict__ pwc,
                                const int* __restrict__ phi,
                                float* __restrict__ out) {
    const int hc = phc[0], wc = pwc[0];
    const int hw = hc * wc;
    const float s_i2c = (float)((double)hc / (double)phi[0]);
    const long long nc = (long long)pB[0] * hw;
    const unsigned long long* __restrict__ win0 = winBase;
    const unsigned long long* __restrict__ win1 = winBase + nc;

    int i = blockIdx.x * blockDim.x + threadIdx.x;
    if (i >= M) return;

    const int b = nt_load_i(mbids, i);
    // s_i2o = scale[:, ::-1]: column x uses scale[b,1], column y uses scale[b,0]
    const float s0x = scale0[2 * b + 1], s0y = scale0[2 * b + 0];
    const float s1x = scale1[2 * b + 1], s1y = scale1[2 * b + 0];

    v2f p0 = nt_load_v2f(mk0, i);                          // single-use stream: TH_NT
    v2f p1 = nt_load_v2f(mk1, i);
    // NOTE: reference does NOT clamp on this path.
    int cx0 = (int)rintf(p0.x * (s_i2c / s0x));
    int cy0 = (int)rintf(p0.y * (s_i2c / s0y));
    int cx1 = (int)rintf(p1.x * (s_i2c / s1x));
    int cy1 = (int)rintf(p1.y * (s_i2c / s1y));

    // Winner tables: 4 MB, L2-resident -> regular temporal loads.
    unsigned id0 = ~(unsigned)(win0[(long long)b * hw + (long long)cy0 * wc + cx0]);
    unsigned id1 = ~(unsigned)(win1[(long long)b * hw + (long long)cy1 * wc + cx1]);

    const bool valid = (id0 < M0) && (id1 < M1);   // unsigned: empty-cell 0xFFFFFFFF fails
    const unsigned id0s = min(id0, M0 - 1u);
    const unsigned id1s = min(id1, M1 - 1u);

    // det_kpts tables (16 MB each) stay hot in the 192 MB L2 -> temporal gathers.
    v2f k0 = *(((const v2f*)dk0) + id0s);
    v2f k1 = *(((const v2f*)dk1) + id1s);

    const float vf = valid ? 1.0f : 0.0f;

    // Tuple outputs, flat in return order: valid, bids, conf, mkpts0_f, mkpts1_f.
    // Write-once streams -> TH_NT stores.
    __builtin_nontemporal_store(vf, out + i);
    __builtin_nontemporal_store(valid ? (float)b : -1.0f, out + (long long)M + i);
    __builtin_nontemporal_store(valid ? nt_load_f(mconf, i) : 0.0f, out + 2LL * M + i);
    v2f o0 = {k0.x * s0x * vf, k0.y * s0y * vf};
    v2f o1 = {k1.x * s1x * vf, k1.y * s1y * vf};
    __builtin_nontemporal_store(o0, ((v2f*)(out + 3LL * M)) + i);
    __builtin_nontemporal_store(o1, ((v2f*)(out + 5LL * M)) + i);
}

extern "C" void kernel_launch(void* const* d_in, const int* in_sizes, int n_in,
                              void* d_out, int out_size, void* d_ws, size_t ws_size,
                              hipStream_t stream) {
    // setup_inputs() order:
    const float* mk0    = (const float*)d_in[0];   // mkpts0_c  (M,2)
    const float* mk1    = (const float*)d_in[1];   // mkpts1_c  (M,2)
    const float* mconf  = (const float*)d_in[2];   // mconf     (M)
    const float* dk0    = (const float*)d_in[3];   // det_kpts0 (M0,2)
    const float* ds0    = (const float*)d_in[4];   // det_scores0 (M0)
    const float* dk1    = (const float*)d_in[5];   // det_kpts1 (M1,2)
    const float* ds1    = (const float*)d_in[6];   // det_scores1 (M1)
    const float* scale0 = (const float*)d_in[7];   // (B,2)
    const float* scale1 = (const float*)d_in[8];   // (B,2)
    const int*   mbids  = (const int*)d_in[9];     // (M)
    const int*   dbids0 = (const int*)d_in[10];    // (M0)
    const int*   dbids1 = (const int*)d_in[11];    // (M1)
    const int*   pB     = (const int*)d_in[12];
    const int*   phc    = (const int*)d_in[13];
    const int*   pwc    = (const int*)d_in[14];
    const int*   phi    = (const int*)d_in[15];

    const int M  = in_sizes[2];
    const int M0 = in_sizes[4];
    const int M1 = in_sizes[6];

    unsigned long long* win = (unsigned long long*)d_ws;  // 2 * B*hc*wc u64 (4 MB)

    const int T = 256;

    // Phase A: zero winner tables (count read on device; fixed grid, grid-stride).
    gm_init_kernel<<<1024, T, 0, stream>>>(win, pB, phc, pwc);

    // Phase B: segmented top-1 via u64 atomic max.
    gm_top1_kernel<<<(M0 + T - 1) / T, T, 0, stream>>>(dk0, ds0, dbids0, M0, 0,
                                                       win, pB, phc, pwc, phi);
    gm_top1_kernel<<<(M1 + T - 1) / T, T, 0, stream>>>(dk1, ds1, dbids1, M1, 1,
                                                       win, pB, phc, pwc, phi);

    // Phase C: per-match lookup/gather/output.
    gm_match_kernel<<<(M + T - 1) / T, T, 0, stream>>>(mk0, mk1, mconf, dk0, dk1,
                                                       scale0, scale1, mbids, win,
                                                       M, (unsigned)M0, (unsigned)M1,
                                                       pB, phc, pwc, phi,
                                                       (float*)d_out);
}